// MarchingTetrahedraHelper_16947940950056
// MI455X (gfx1250) — compile-verified
//
#include <hip/hip_runtime.h>
#include <cstdint>
#include <cmath>

#ifndef USE_TDM
#define USE_TDM 1
#endif

// ---------------------------------------------------------------------------
// Constant tables (marching tetrahedra)
// ---------------------------------------------------------------------------
__constant__ int TRI_TAB[16][6] = {
    {-1,-1,-1,-1,-1,-1},{1,0,2,-1,-1,-1},{4,0,3,-1,-1,-1},{1,4,2,1,3,4},
    {3,1,5,-1,-1,-1},{2,3,0,2,5,3},{1,4,0,1,5,4},{4,2,5,-1,-1,-1},
    {4,5,2,-1,-1,-1},{4,1,0,4,5,1},{3,2,0,3,5,2},{1,3,5,-1,-1,-1},
    {4,1,2,4,3,1},{3,0,4,-1,-1,-1},{2,0,1,-1,-1,-1},{-1,-1,-1,-1,-1,-1}};
__constant__ int TET_C[6][4] = {{0,1,3,7},{0,3,2,7},{0,2,6,7},{0,6,4,7},{0,4,5,7},{0,5,1,7}};
__constant__ int EDG_A[6] = {0,0,0,1,1,2};
__constant__ int EDG_B[6] = {1,2,3,2,3,3};

// ---------------------------------------------------------------------------
// CDNA5 Tensor Data Mover: 2D tile (dim0 contiguous, dim1 strided) -> LDS.
// D# packed per CDNA5 ISA ch.8: group0 (4 SGPRs), group1 (8 SGPRs).
// dsz_code: 0=1B, 1=2B, 2=4B elements. All dims in elements.
// ---------------------------------------------------------------------------
typedef uint32_t u32x4 __attribute__((ext_vector_type(4)));
typedef uint32_t u32x8 __attribute__((ext_vector_type(8)));

__device__ __forceinline__ void tdm_load_2d_to_lds(
    uint32_t lds_byte, const void* gptr, uint32_t dsz_code,
    uint32_t tensor_d0, uint32_t tensor_d1,
    uint32_t tile_d0, uint32_t tile_d1, uint32_t stride0_elems)
{
  uint64_t ga = (uint64_t)(uintptr_t)gptr;
  u32x4 g0;
  g0[0] = 1u;                                                  // count=1 (valid), user mode
  g0[1] = lds_byte;                                            // lds_addr (bytes)
  g0[2] = (uint32_t)ga;                                        // global_addr[31:0]
  g0[3] = ((uint32_t)(ga >> 32) & 0x01FFFFFFu) | (2u << 30);   // addr[56:32] | type=2
  u32x8 g1;
  g1[0] = (dsz_code & 3u) << 16;                               // wg_mask=0 (not in cluster), data_size
  g1[1] = (tensor_d0 & 0xFFFFu) << 16;                         // tensor_dim0[15:0] @ bits63:48
  g1[2] = ((tensor_d0 >> 16) & 0xFFFFu) | ((tensor_d1 & 0xFFFFu) << 16);
  g1[3] = ((tensor_d1 >> 16) & 0xFFFFu) | ((tile_d0 & 0xFFFFu) << 16);
  g1[4] = (tile_d1 & 0xFFFFu);                                 // tile_dim2 = 0 (2D tile)
  g1[5] = stride0_elems;                                       // tensor_dim0_stride[31:0]
  g1[6] = 0u;
  g1[7] = 0u;
  asm volatile("tensor_load_to_lds %0, %1" :: "s"(g0), "s"(g1) : "memory");
}

// ---------------------------------------------------------------------------
// Kernel 1: pos = grid_vertices + tanh(deformation)/R
// ---------------------------------------------------------------------------
__global__ void mt_pos_kernel(const float* __restrict__ gv, const float* __restrict__ defo,
                              float* __restrict__ pos, int n3, float Rf)
{
  int i = blockIdx.x * blockDim.x + threadIdx.x;
  if (i < n3) pos[i] = gv[i] + tanhf(defo[i]) / Rf;
}

// ---------------------------------------------------------------------------
// Kernel 2: per-vertex 7-bit crossing-edge mask + count.
// Edge offsets in ascending global-id order: 1, r1, r1+1, r1^2, r1^2+1, r1^2+r1, r1^2+r1+1
// ---------------------------------------------------------------------------
__global__ void mt_edge_mask_kernel(const float* __restrict__ level, uint8_t* __restrict__ mask,
                                    uint32_t* __restrict__ cnt, int Nv, int r1, int R)
{
  int v = blockIdx.x * blockDim.x + threadIdx.x;
  if (v >= Nv) return;
  int r1s = r1 * r1;
  int i = v / r1s; int rem = v - i * r1s;
  int j = rem / r1; int k = rem - j * r1;
  bool bi = i < R, bj = j < R, bk = k < R;
  float l0 = level[v]; bool o0 = l0 > 0.0f;
  int D[7]  = {1, r1, r1 + 1, r1s, r1s + 1, r1s + r1, r1s + r1 + 1};
  bool B[7] = {bk, bj, bj && bk, bi, bi && bk, bi && bj, bi && bj && bk};
  uint32_t m = 0;
  #pragma unroll
  for (int o = 0; o < 7; ++o) {
    if (B[o]) {
      bool ob = level[v + D[o]] > 0.0f;
      if (ob != o0) m |= (1u << o);
    }
  }
  mask[v] = (uint8_t)m;
  cnt[v] = (uint32_t)__popc(m);
}

// ---------------------------------------------------------------------------
// Hierarchical exclusive scan (1024 items/block = 256 threads x 4)
// ---------------------------------------------------------------------------
__global__ void mt_scan_block_u32(const uint32_t* __restrict__ in, uint32_t* __restrict__ out,
                                  uint32_t* __restrict__ bsum, int n)
{
  __shared__ uint32_t sh[256];
  int t = threadIdx.x;
  int base = blockIdx.x * 1024 + t * 4;
  uint32_t v0 = (base + 0 < n) ? in[base + 0] : 0u;
  uint32_t v1 = (base + 1 < n) ? in[base + 1] : 0u;
  uint32_t v2 = (base + 2 < n) ? in[base + 2] : 0u;
  uint32_t v3 = (base + 3 < n) ? in[base + 3] : 0u;
  uint32_t p1 = v0, p2 = v0 + v1, p3 = v0 + v1 + v2;
  uint32_t tsum = p3 + v3;
  sh[t] = tsum;
  __syncthreads();
  for (int off = 1; off < 256; off <<= 1) {
    uint32_t x = (t >= off) ? sh[t - off] : 0u;
    __syncthreads();
    sh[t] += x;
    __syncthreads();
  }
  uint32_t excl = sh[t] - tsum;
  if (base + 0 < n) out[base + 0] = excl;
  if (base + 1 < n) out[base + 1] = excl + p1;
  if (base + 2 < n) out[base + 2] = excl + p2;
  if (base + 3 < n) out[base + 3] = excl + p3;
  if (t == 0) bsum[blockIdx.x] = sh[255];
}

__global__ void mt_scan_partials_u32(uint32_t* __restrict__ bs, int nb, uint32_t* __restrict__ total)
{
  __shared__ uint32_t sh[256];
  __shared__ uint32_t carrySh;
  int t = threadIdx.x;
  if (t == 0) carrySh = 0;
  __syncthreads();
  for (int start = 0; start < nb; start += 256) {
    int i = start + t;
    uint32_t v = (i < nb) ? bs[i] : 0u;
    sh[t] = v;
    __syncthreads();
    for (int off = 1; off < 256; off <<= 1) {
      uint32_t x = (t >= off) ? sh[t - off] : 0u;
      __syncthreads();
      sh[t] += x;
      __syncthreads();
    }
    uint32_t c = carrySh;
    if (i < nb) bs[i] = c + sh[t] - v;
    uint32_t tot = sh[255];
    __syncthreads();
    if (t == 0) carrySh = c + tot;
    __syncthreads();
  }
  if (t == 0) *total = carrySh;
}

__global__ void mt_add_back_u32(uint32_t* __restrict__ d, const uint32_t* __restrict__ bs, int n)
{
  int i = blockIdx.x * blockDim.x + threadIdx.x;
  if (i < n) d[i] += bs[i >> 10];
}

// ---------------------------------------------------------------------------
// Kernel: per-tet triangle-count predicates (popcnt 1/3 -> 1 tri, 2 -> 2 tris)
// ---------------------------------------------------------------------------
__global__ void mt_tet_count_kernel(const float* __restrict__ level, const int* __restrict__ tets,
                                    uint32_t* __restrict__ c1, uint32_t* __restrict__ c2, int F)
{
  int t = blockIdx.x * blockDim.x + threadIdx.x;
  if (t >= F) return;
  int4 q = ((const int4*)tets)[t];
  int s = (level[q.x] > 0.0f) + (level[q.y] > 0.0f) + (level[q.z] > 0.0f) + (level[q.w] > 0.0f);
  c1[t] = (s == 1 || s == 3) ? 1u : 0u;
  c2[t] = (s == 2) ? 1u : 0u;
}

// ---------------------------------------------------------------------------
// Kernel: emit interpolated crossing-edge vertices (sorted (v, delta) order)
// ---------------------------------------------------------------------------
__global__ void mt_edge_emit_kernel(const float* __restrict__ level, const float* __restrict__ pos,
                                    const uint8_t* __restrict__ mask, const uint32_t* __restrict__ ebase,
                                    float* __restrict__ out, int Nv, int r1)
{
  int v = blockIdx.x * blockDim.x + threadIdx.x;
  if (v >= Nv) return;
  uint32_t m = mask[v];
  if (!m) return;
  int r1s = r1 * r1;
  int D[7] = {1, r1, r1 + 1, r1s, r1s + 1, r1s + r1, r1s + r1 + 1};
  uint32_t idx = ebase[v];
  float la = level[v];
  float ax = pos[3 * v + 0], ay = pos[3 * v + 1], az = pos[3 * v + 2];
  #pragma unroll
  for (int o = 0; o < 7; ++o) {
    if ((m >> o) & 1u) {
      int b = v + D[o];
      float lb = level[b];
      float denom = la - lb;                 // sdf0 - sdf1
      float w0 = -lb / denom, w1 = la / denom;
      size_t p = (size_t)3 * idx;
      out[p + 0] = ax * w0 + pos[3 * b + 0] * w1;
      out[p + 1] = ay * w0 + pos[3 * b + 1] * w1;
      out[p + 2] = az * w0 + pos[3 * b + 2] * w1;
      ++idx;
    }
  }
}

// ---------------------------------------------------------------------------
// Face kernel: each block handles a brick of 4(j) x 64(k) cubes at fixed i.
// Needed data (level / edgeBase / edgeMask over a 2 x 5 x 65 vertex brick) is
// DMA-staged into LDS with the CDNA5 Tensor Data Mover (wave 0 issues 6 TDM
// loads, waits TENSORcnt, then the whole group barriers). Assumes r1 <= 65.
// ---------------------------------------------------------------------------
__global__ void mt_faces_kernel(const float* __restrict__ level,
                                const uint32_t* __restrict__ ebase,
                                const uint8_t* __restrict__ emask,
                                const uint32_t* __restrict__ tc1,
                                const uint32_t* __restrict__ tc2,
                                const uint32_t* __restrict__ totals,
                                float* __restrict__ out,
                                int r1, int R, int jg)
{
  __shared__ float    sLevel[650];
  __shared__ uint32_t sBase[650];
  __shared__ uint8_t  sMask[656];

  const int ci = blockIdx.x / jg;
  const int j0 = (blockIdx.x % jg) * 4;
  const int lslice = 5 * r1;
  const int mslice = (5 * r1 + 7) & ~7;
  const int jext = (5 < r1 - j0) ? 5 : (r1 - j0);

#if USE_TDM
  if (threadIdx.x < 32u) {
    #pragma unroll
    for (int li = 0; li < 2; ++li) {
      size_t vtx = ((size_t)(ci + li) * r1 + j0) * (size_t)r1;
      tdm_load_2d_to_lds((uint32_t)(uintptr_t)&sLevel[li * lslice], level + vtx, 2u,
                         (uint32_t)r1, (uint32_t)r1, (uint32_t)r1, (uint32_t)jext, (uint32_t)r1);
      tdm_load_2d_to_lds((uint32_t)(uintptr_t)&sBase[li * lslice], ebase + vtx, 2u,
                         (uint32_t)r1, (uint32_t)r1, (uint32_t)r1, (uint32_t)jext, (uint32_t)r1);
      tdm_load_2d_to_lds((uint32_t)(uintptr_t)&sMask[li * mslice], emask + vtx, 0u,
                         (uint32_t)r1, (uint32_t)r1, (uint32_t)r1, (uint32_t)jext, (uint32_t)r1);
    }
    __builtin_amdgcn_s_wait_tensorcnt(0);
  }
  __syncthreads();
#else
  for (int idx = threadIdx.x; idx < 2 * lslice; idx += blockDim.x) {
    int li = idx / lslice; int rem = idx - li * lslice;
    int lj = rem / r1;     int lk2 = rem - lj * r1;
    if (lj < jext) {
      size_t g = ((size_t)(ci + li) * r1 + (j0 + lj)) * (size_t)r1 + lk2;
      sLevel[idx] = level[g];
      sBase[idx]  = ebase[g];
      sMask[li * mslice + lj * r1 + lk2] = emask[g];
    }
  }
  __syncthreads();
#endif

  const int lj4 = threadIdx.x >> 6;   // 0..3
  const int lk  = threadIdx.x & 63;   // 0..63
  const int cj = j0 + lj4;
  const int ck = lk;
  if (cj >= R || ck >= R) return;

  int dG[8], lOf[8], mOf[8];
  #pragma unroll
  for (int c = 0; c < 8; ++c) {
    int dx = (c >> 2) & 1, dy = (c >> 1) & 1, dz = c & 1;
    dG[c]  = (dx * r1 + dy) * r1 + dz;
    lOf[c] = dx * lslice + (lj4 + dy) * r1 + (lk + dz);
    mOf[c] = dx * mslice + (lj4 + dy) * r1 + (lk + dz);
  }
  const uint32_t M  = totals[0];
  const uint32_t N1 = totals[1];
  const size_t faceBase = (size_t)3 * M;
  const int cubeFlat = (ci * R + cj) * R + ck;
  const int r1s = r1 * r1;

  for (int tt = 0; tt < 6; ++tt) {
    int cs0 = TET_C[tt][0], cs1 = TET_C[tt][1], cs2 = TET_C[tt][2], cs3 = TET_C[tt][3];
    int o0 = sLevel[lOf[cs0]] > 0.0f;
    int o1 = sLevel[lOf[cs1]] > 0.0f;
    int o2 = sLevel[lOf[cs2]] > 0.0f;
    int o3 = sLevel[lOf[cs3]] > 0.0f;
    int pc = o0 + o1 + o2 + o3;
    if (pc == 0 || pc == 4) continue;
    int ti = o0 + o1 * 2 + o2 * 4 + o3 * 8;
    int cs[4] = {cs0, cs1, cs2, cs3};
    int im[6];
    #pragma unroll
    for (int e = 0; e < 6; ++e) {
      int ca = cs[EDG_A[e]], cb = cs[EDG_B[e]];
      int cmin = ca < cb ? ca : cb;
      int cmax = ca < cb ? cb : ca;
      int d = dG[cmax] - dG[cmin];
      int cls = (d == 1) ? 0 : (d == r1) ? 1 : (d == r1 + 1) ? 2 : (d == r1s) ? 3
              : (d == r1s + 1) ? 4 : (d == r1s + r1) ? 5 : 6;
      uint32_t mk = sMask[mOf[cmin]];
      im[e] = (int)(sBase[lOf[cmin]] + (uint32_t)__popc(mk & ((1u << cls) - 1u)));
    }
    int t = cubeFlat * 6 + tt;
    const int* row = TRI_TAB[ti];
    if (pc == 2) {
      uint32_t r0 = N1 + 2u * tc2[t];
      size_t o = faceBase + (size_t)3 * r0;
      out[o + 0] = (float)im[row[0]]; out[o + 1] = (float)im[row[1]]; out[o + 2] = (float)im[row[2]];
      out[o + 3] = (float)im[row[3]]; out[o + 4] = (float)im[row[4]]; out[o + 5] = (float)im[row[5]];
    } else {
      uint32_t r = tc1[t];
      size_t o = faceBase + (size_t)3 * r;
      out[o + 0] = (float)im[row[0]]; out[o + 1] = (float)im[row[1]]; out[o + 2] = (float)im[row[2]];
    }
  }
}

// ---------------------------------------------------------------------------
// Host launcher
// ---------------------------------------------------------------------------
extern "C" void kernel_launch(void* const* d_in, const int* in_sizes, int n_in,
                              void* d_out, int out_size, void* d_ws, size_t ws_size,
                              hipStream_t stream)
{
  const float* level = (const float*)d_in[0];
  const float* defo  = (const float*)d_in[1];
  const float* gv    = (const float*)d_in[2];
  const int*   tets  = (const int*)d_in[3];

  const int Nv = in_sizes[0];
  const int r1 = (int)llround(cbrt((double)Nv));
  const int R  = r1 - 1;
  const int F  = in_sizes[3] / 4;

  // Carve workspace (256B aligned chunks)
  size_t off = 0;
  char* w = (char*)d_ws;
  auto carve = [&](size_t bytes) -> void* {
    off = (off + 255) & ~(size_t)255;
    void* p = w + off;
    off += bytes;
    return p;
  };
  float*    pos    = (float*)   carve((size_t)Nv * 3 * sizeof(float));
  uint32_t* ebase  = (uint32_t*)carve((size_t)Nv * sizeof(uint32_t));
  uint8_t*  emask  = (uint8_t*) carve((size_t)Nv);
  uint32_t* tc1    = (uint32_t*)carve((size_t)F * sizeof(uint32_t));
  uint32_t* tc2    = (uint32_t*)carve((size_t)F * sizeof(uint32_t));
  const int nvB = (Nv + 1023) / 1024;
  const int ftB = (F + 1023) / 1024;
  const int bsN = (nvB > ftB ? nvB : ftB) + 1;
  uint32_t* bsum   = (uint32_t*)carve((size_t)bsN * sizeof(uint32_t));
  uint32_t* totals = (uint32_t*)carve(4 * sizeof(uint32_t));

  const int B = 256;
  // 1. deformed positions
  mt_pos_kernel<<<(3 * Nv + B - 1) / B, B, 0, stream>>>(gv, defo, pos, 3 * Nv, (float)R);
  // 2. crossing-edge masks + counts
  mt_edge_mask_kernel<<<(Nv + B - 1) / B, B, 0, stream>>>(level, emask, ebase, Nv, r1, R);
  // 3. exclusive scan of edge counts -> compact edge indices, total M -> totals[0]
  mt_scan_block_u32<<<nvB, B, 0, stream>>>(ebase, ebase, bsum, Nv);
  mt_scan_partials_u32<<<1, B, 0, stream>>>(bsum, nvB, &totals[0]);
  mt_add_back_u32<<<(Nv + B - 1) / B, B, 0, stream>>>(ebase, bsum, Nv);
  // 4. per-tet triangle-count predicates
  mt_tet_count_kernel<<<(F + B - 1) / B, B, 0, stream>>>(level, tets, tc1, tc2, F);
  // 5. scan tc1 (1-tri tets) -> totals[1] = N1
  mt_scan_block_u32<<<ftB, B, 0, stream>>>(tc1, tc1, bsum, F);
  mt_scan_partials_u32<<<1, B, 0, stream>>>(bsum, ftB, &totals[1]);
  mt_add_back_u32<<<(F + B - 1) / B, B, 0, stream>>>(tc1, bsum, F);
  // 6. scan tc2 (2-tri tets) -> totals[2]
  mt_scan_block_u32<<<ftB, B, 0, stream>>>(tc2, tc2, bsum, F);
  mt_scan_partials_u32<<<1, B, 0, stream>>>(bsum, ftB, &totals[2]);
  mt_add_back_u32<<<(F + B - 1) / B, B, 0, stream>>>(tc2, bsum, F);
  // 7. emit interpolated vertices into d_out[0 .. 3M)
  mt_edge_emit_kernel<<<(Nv + B - 1) / B, B, 0, stream>>>(level, pos, emask, ebase,
                                                          (float*)d_out, Nv, r1);
  // 8. emit faces into d_out[3M .. 3M+3Nf) (TDM-staged bricks)
  const int jg = (R + 3) / 4;
  mt_faces_kernel<<<R * jg, B, 0, stream>>>(level, ebase, emask, tc1, tc2, totals,
                                            (float*)d_out, r1, R, jg);
}